// CRF_Tagger_73435350827400
// MI455X (gfx1250) — compile-verified
//
#include <hip/hip_runtime.h>
#include <hip/hip_bf16.h>
#include <math.h>

// Problem constants (match reference).
#define BB 64
#define TT 512
#define DD 768
#define KK 32

typedef __attribute__((ext_vector_type(2))) float v2f;
typedef __attribute__((ext_vector_type(8))) float v8f;

// -----------------------------------------------------------------------------
// Kernel 0: Wt[n][d] = W[d][n]  (32 x 768 transpose, one-time, 96 KB)
// Makes each lane's WMMA B-operand pair (W[kk][n], W[kk+1][n]) contiguous in
// memory -> single b64 load per B tile per chunk.
// -----------------------------------------------------------------------------
__global__ __launch_bounds__(256)
void transpose_W(const float* __restrict__ W, float* __restrict__ Wt) {
    for (int i = threadIdx.x + blockIdx.x * 256; i < DD * KK; i += 256 * 96) {
        const int d = i / KK;
        const int n = i % KK;
        Wt[(size_t)n * DD + d] = W[i];
    }
}

// -----------------------------------------------------------------------------
// Kernel 1: emissions = x @ W + b   using V_WMMA_F32_16X16X4_F32
//   x: [B*T, D] row-major, Wt: [K, D] (transposed W), b: [K]
//   Each wave computes a 32-row x 32-col tile: 2 row slabs x 2 col tiles,
//   looping D in chunks of 4 (the fp32 WMMA K), register double-buffered so
//   loads for chunk d0+4 are in flight while WMMAs for d0 execute.
// A-operand layout (16x4 f32): lanes 0-15 hold rows M=0..15 with K=0,1 in
// VGPR0/1; lanes 16-31 hold the same rows with K=2,3. -> one b64 load/lane.
// B-operand layout (4x16 f32): lane n (0-15) holds B[0][n],B[1][n]; lane n+16
// holds B[2][n],B[3][n]. With Wt this is a contiguous b64 per lane.
// C/D layout (16x16 f32): column n = lane&15; VGPR v gives row v + 8*(lane>>4).
// x is streamed exactly once -> nontemporal loads keep L2 for Wt/emissions.
// -----------------------------------------------------------------------------
__global__ __launch_bounds__(256)
void gemm_emissions_wmma(const float* __restrict__ x,
                         const float* __restrict__ Wt,
                         const float* __restrict__ bias,
                         float* __restrict__ em) {
    const int lane = threadIdx.x & 31;
    const int wave = threadIdx.x >> 5;
    const int m    = lane & 15;      // row-within-slab / B column
    const int half = lane >> 4;      // 0: K=0,1   1: K=2,3

    const int row0 = (blockIdx.x * 8 + wave) * 32;   // 32 rows per wave

    v8f c00 = {}; v8f c01 = {}; v8f c10 = {}; v8f c11 = {};

    const float* __restrict__ xa0 = x + (size_t)(row0 + m) * DD + 2 * half;
    const float* __restrict__ xa1 = x + (size_t)(row0 + 16 + m) * DD + 2 * half;
    const float* __restrict__ wb0 = Wt + (size_t)(m)      * DD + 2 * half;
    const float* __restrict__ wb1 = Wt + (size_t)(m + 16) * DD + 2 * half;

    // Prologue: chunk 0 in registers.
    v2f a0 = __builtin_nontemporal_load((const v2f*)(xa0));
    v2f a1 = __builtin_nontemporal_load((const v2f*)(xa1));
    v2f b0 = *(const v2f*)(wb0);
    v2f b1 = *(const v2f*)(wb1);

    for (int d0 = 0; d0 < DD - 4; d0 += 4) {
        // Issue next chunk's loads before consuming the current one.
        const int dn = d0 + 4;
        v2f na0 = __builtin_nontemporal_load((const v2f*)(xa0 + dn));
        v2f na1 = __builtin_nontemporal_load((const v2f*)(xa1 + dn));
        v2f nb0 = *(const v2f*)(wb0 + dn);
        v2f nb1 = *(const v2f*)(wb1 + dn);

        c00 = __builtin_amdgcn_wmma_f32_16x16x4_f32(false, a0, false, b0,
                                                    (short)0, c00, false, false);
        c01 = __builtin_amdgcn_wmma_f32_16x16x4_f32(false, a0, false, b1,
                                                    (short)0, c01, false, false);
        c10 = __builtin_amdgcn_wmma_f32_16x16x4_f32(false, a1, false, b0,
                                                    (short)0, c10, false, false);
        c11 = __builtin_amdgcn_wmma_f32_16x16x4_f32(false, a1, false, b1,
                                                    (short)0, c11, false, false);
        a0 = na0; a1 = na1; b0 = nb0; b1 = nb1;
    }
    // Epilogue chunk.
    c00 = __builtin_amdgcn_wmma_f32_16x16x4_f32(false, a0, false, b0,
                                                (short)0, c00, false, false);
    c01 = __builtin_amdgcn_wmma_f32_16x16x4_f32(false, a0, false, b1,
                                                (short)0, c01, false, false);
    c10 = __builtin_amdgcn_wmma_f32_16x16x4_f32(false, a1, false, b0,
                                                (short)0, c10, false, false);
    c11 = __builtin_amdgcn_wmma_f32_16x16x4_f32(false, a1, false, b1,
                                                (short)0, c11, false, false);

    const float bv0 = bias[m];
    const float bv1 = bias[m + 16];

    #pragma unroll
    for (int v = 0; v < 8; ++v) {
        const int r = v + 8 * half;
        em[(size_t)(row0 + r     ) * KK + m     ] = c00[v] + bv0;
        em[(size_t)(row0 + r     ) * KK + m + 16] = c01[v] + bv1;
        em[(size_t)(row0 + 16 + r) * KK + m     ] = c10[v] + bv0;
        em[(size_t)(row0 + 16 + r) * KK + m + 16] = c11[v] + bv1;
    }
}

// -----------------------------------------------------------------------------
// Kernel 2: per-batch CRF forward + gold numerator -> llh[b]
//   One wave (32 lanes) per batch; lane = tag k. trans column held in 32 VGPRs.
// -----------------------------------------------------------------------------
__global__ __launch_bounds__(32)
void crf_forward(const float* __restrict__ em,          // [B,T,K]
                 const int* __restrict__ labels,        // [B,T]
                 const unsigned char* __restrict__ mask,// [B,T] (bool)
                 const float* __restrict__ start_trans,
                 const float* __restrict__ end_trans,
                 const float* __restrict__ trans,       // [K,K]
                 float* __restrict__ llh) {             // [B]
    const int b    = blockIdx.x;
    const int lane = threadIdx.x;

    // trans[:, lane] resident in registers.
    float tr[KK];
    #pragma unroll
    for (int j = 0; j < KK; ++j) tr[j] = trans[j * KK + lane];

    const size_t emb = (size_t)b * TT * KK;
    const int    lb  = b * TT;

    // ---- gold-path numerator (lane-parallel over t, then wave reduce) ----
    float numacc = 0.0f;
    for (int t = 1 + lane; t < TT; t += 32) {
        const int p  = labels[lb + t - 1];
        const int c  = labels[lb + t];
        const float mf = mask[lb + t] ? 1.0f : 0.0f;
        numacc += (trans[p * KK + c] + em[emb + (size_t)t * KK + c]) * mf;
    }
    int cnt = 0;
    for (int t = lane; t < TT; t += 32) cnt += mask[lb + t] ? 1 : 0;
    #pragma unroll
    for (int off = 16; off > 0; off >>= 1) {
        numacc += __shfl_xor(numacc, off, 32);
        cnt    += __shfl_xor(cnt,    off, 32);
    }

    // ---- forward recursion ----
    __shared__ float sLDS[KK];
    float s = start_trans[lane] + em[emb + lane];

    for (int t = 1; t < TT; ++t) {
        sLDS[lane] = s;
        __syncthreads();
        float v[KK];
        float mx = -INFINITY;
        #pragma unroll
        for (int j = 0; j < KK; ++j) {
            v[j] = sLDS[j] + tr[j];
            mx = fmaxf(mx, v[j]);
        }
        float sum = 0.0f;
        #pragma unroll
        for (int j = 0; j < KK; ++j) sum += __expf(v[j] - mx);
        const float nxt = mx + __logf(sum) + em[emb + (size_t)t * KK + lane];
        s = mask[lb + t] ? nxt : s;
        __syncthreads();
    }

    // ---- denominator: logsumexp over lanes of s + end_trans ----
    float sf = s + end_trans[lane];
    float mx = sf;
    #pragma unroll
    for (int off = 16; off > 0; off >>= 1)
        mx = fmaxf(mx, __shfl_xor(mx, off, 32));
    float ex = __expf(sf - mx);
    #pragma unroll
    for (int off = 16; off > 0; off >>= 1)
        ex += __shfl_xor(ex, off, 32);

    if (lane == 0) {
        const float denom = mx + __logf(ex);
        const int l0   = labels[lb];
        const int last = labels[lb + cnt - 1];
        const float num = numacc + start_trans[l0] + em[emb + l0] + end_trans[last];
        llh[b] = num - denom;
    }
}

// -----------------------------------------------------------------------------
// Kernel 3: out[0] = -mean(llh)
// -----------------------------------------------------------------------------
__global__ __launch_bounds__(32)
void reduce_mean_neg(const float* __restrict__ llh, float* __restrict__ out) {
    const int lane = threadIdx.x;
    float v = llh[lane] + llh[lane + 32];
    #pragma unroll
    for (int off = 16; off > 0; off >>= 1)
        v += __shfl_xor(v, off, 32);
    if (lane == 0) out[0] = -(v / (float)BB);
}

// -----------------------------------------------------------------------------
// Launch
// -----------------------------------------------------------------------------
extern "C" void kernel_launch(void* const* d_in, const int* in_sizes, int n_in,
                              void* d_out, int out_size, void* d_ws, size_t ws_size,
                              hipStream_t stream) {
    const float*         x     = (const float*)d_in[0];
    const int*           lab   = (const int*)d_in[1];
    const unsigned char* mask  = (const unsigned char*)d_in[2];
    const float*         W     = (const float*)d_in[3];
    const float*         bias  = (const float*)d_in[4];
    const float*         st    = (const float*)d_in[5];
    const float*         et    = (const float*)d_in[6];
    const float*         trans = (const float*)d_in[7];

    float* em  = (float*)d_ws;                       // B*T*K floats = 4 MB
    float* llh = em + (size_t)BB * TT * KK;          // 64 floats
    float* Wt  = llh + BB;                           // K*D floats = 96 KB

    // One-time W transpose (tiny).
    transpose_W<<<dim3(96), dim3(256), 0, stream>>>(W, Wt);

    // GEMM: 32768 rows / (32 rows/wave * 8 waves/block) = 128 blocks.
    gemm_emissions_wmma<<<dim3(128), dim3(256), 0, stream>>>(x, Wt, bias, em);

    // CRF forward: one wave per batch.
    crf_forward<<<dim3(BB), dim3(32), 0, stream>>>(em, lab, mask, st, et, trans, llh);

    // Final scalar.
    reduce_mean_neg<<<dim3(1), dim3(32), 0, stream>>>(llh, (float*)d_out);
}